// ResidualConvBlock_47742856463189
// MI455X (gfx1250) — compile-verified
//
#include <hip/hip_runtime.h>
#include <math.h>

#define DFEAT 128

typedef float v2f __attribute__((ext_vector_type(2)));
typedef float v8f __attribute__((ext_vector_type(8)));

__device__ __forceinline__ v8f wmma4(v2f a, v2f b, v8f c) {
  // D(16x16,f32) = A(16x4,f32) x B(4x16,f32) + C   -> v_wmma_f32_16x16x4_f32
  return __builtin_amdgcn_wmma_f32_16x16x4_f32(
      /*neg_a=*/false, a, /*neg_b=*/false, b,
      /*c_mod=*/(short)0, c, /*reuse_a=*/false, /*reuse_b=*/false);
}

// ---------------------------------------------------------------- zero fill
__global__ void zero_f32(float* __restrict__ p, long n) {
  long i = (long)blockIdx.x * blockDim.x + threadIdx.x;
  if (i < n) p[i] = 0.0f;
}

// ---------------------------------------------------------------- LayerNorm1
// one wave32 per row, 4 floats per lane, shuffle reductions (wave32!)
__global__ void ln1_kernel(const float* __restrict__ h,
                           const float* __restrict__ g,
                           const float* __restrict__ b,
                           float* __restrict__ out, int N) {
  int wave = threadIdx.x >> 5;
  int lane = threadIdx.x & 31;
  int row  = blockIdx.x * (blockDim.x >> 5) + wave;
  if (row >= N) return;
  float4 v = reinterpret_cast<const float4*>(h + (size_t)row * DFEAT)[lane];
  float s = v.x + v.y + v.z + v.w;
#pragma unroll
  for (int m = 16; m >= 1; m >>= 1) s += __shfl_xor(s, m, 32);
  float mu = s * (1.0f / DFEAT);
  float dx = v.x - mu, dy = v.y - mu, dz = v.z - mu, dw = v.w - mu;
  float q = dx * dx + dy * dy + dz * dz + dw * dw;
#pragma unroll
  for (int m = 16; m >= 1; m >>= 1) q += __shfl_xor(q, m, 32);
  float inv = rsqrtf(q * (1.0f / DFEAT) + 1e-5f);
  float4 gv = reinterpret_cast<const float4*>(g)[lane];
  float4 bv = reinterpret_cast<const float4*>(b)[lane];
  float4 o;
  o.x = dx * inv * gv.x + bv.x;
  o.y = dy * inv * gv.y + bv.y;
  o.z = dz * inv * gv.z + bv.z;
  o.w = dw * inv * gv.w + bv.w;
  reinterpret_cast<float4*>(out + (size_t)row * DFEAT)[lane] = o;
}

// ---------------------------------------------------------------- degree
__global__ void deg_kernel(const int* __restrict__ dst, float* __restrict__ deg, int E) {
  int e = blockIdx.x * blockDim.x + threadIdx.x;
  if (e < E) atomicAdd(&deg[dst[e]], 1.0f);
}

// ---------------------------------------------------------------- edge scatter
// 32 lanes per edge, float4 gather from h_norm[src], f32 atomic adds into agg[dst]
__global__ void scatter_kernel(const float* __restrict__ hn,
                               const int* __restrict__ src,
                               const int* __restrict__ dst,
                               float* __restrict__ agg, int E) {
  long gid = (long)blockIdx.x * blockDim.x + threadIdx.x;
  long e = gid >> 5;
  int lane = (int)(gid & 31);
  if (e >= E) return;
  int s = src[e], d = dst[e];
  float4 v = reinterpret_cast<const float4*>(hn + (size_t)s * DFEAT)[lane];
  float* a = agg + (size_t)d * DFEAT + lane * 4;
  atomicAdd(a + 0, v.x);
  atomicAdd(a + 1, v.y);
  atomicAdd(a + 2, v.z);
  atomicAdd(a + 3, v.w);
}

// ---------------------------------------------------------------- fused tail
// per wave32: 16 rows. conv = X1*Wself^T + (agg/deg)*Wneigh^T + b + X1 (fp32 WMMA),
// in-register LN2, stage via LDS, WMMA GEMM with Wsi, ELU + residual.
__global__ void __launch_bounds__(32)
fused_conv_kernel(const float* __restrict__ X1,
                  const float* __restrict__ agg,
                  const float* __restrict__ deg,
                  const float* __restrict__ Wself,
                  const float* __restrict__ Wneigh,
                  const float* __restrict__ bconv,
                  const float* __restrict__ g2,
                  const float* __restrict__ b2,
                  const float* __restrict__ Wsi,
                  const float* __restrict__ bsi,
                  float* __restrict__ out, int N) {
  __shared__ float lds[16 * 132];

  const int lane = threadIdx.x;       // 0..31
  const int half = lane >> 4;         // 0: K={0,1}/rows M..M+7 ; 1: K={2,3}/rows M+8..
  const int l16  = lane & 15;
  const int i0   = blockIdx.x * 16;
  const int kq   = half * 2;

  int rA = i0 + l16;                  // A-fragment row this lane loads
  if (rA > N - 1) rA = N - 1;         // clamp (keeps EXEC full for WMMA)
  const float invd = 1.0f / fmaxf(deg[rA], 1.0f);

  const v8f vzero = {0.f, 0.f, 0.f, 0.f, 0.f, 0.f, 0.f, 0.f};
  v8f acc[8];
#pragma unroll
  for (int t = 0; t < 8; ++t) acc[t] = vzero;

  const float* x1row  = X1  + (size_t)rA * DFEAT + kq;
  const float* aggrow = agg + (size_t)rA * DFEAT + kq;

  // ---- GEMM1+2: conv accumulation, K = 128 in 32 steps of 4
#pragma unroll 4
  for (int kk = 0; kk < 32; ++kk) {
    const int k0 = kk * 4;
    v2f aS = *reinterpret_cast<const v2f*>(x1row + k0);
    v2f aN = *reinterpret_cast<const v2f*>(aggrow + k0);
    aN *= invd;                                  // h_neigh = agg / max(deg,1)
#pragma unroll
    for (int t = 0; t < 8; ++t) {
      const int colB = t * 16 + l16;             // output column (row of W)
      v2f bS = *reinterpret_cast<const v2f*>(Wself  + (size_t)colB * DFEAT + k0 + kq);
      v2f bN = *reinterpret_cast<const v2f*>(Wneigh + (size_t)colB * DFEAT + k0 + kq);
      acc[t] = wmma4(aS, bS, acc[t]);
      acc[t] = wmma4(aN, bN, acc[t]);
    }
  }

  // per-lane column-constant params for the 8 tiles
  float bc[8], gl[8], bl[8], bs[8];
#pragma unroll
  for (int t = 0; t < 8; ++t) {
    int c = t * 16 + l16;
    bc[t] = bconv[c]; gl[t] = g2[c]; bl[t] = b2[c]; bs[t] = bsi[c];
  }

  // ---- bias + residual + LayerNorm2 (rows live in one 16-lane half)
#pragma unroll
  for (int r = 0; r < 8; ++r) {
    int rowR = i0 + r + half * 8;
    int rowC = rowR > N - 1 ? N - 1 : rowR;
    float vals[8];
    float s = 0.f;
#pragma unroll
    for (int t = 0; t < 8; ++t) {
      int c = t * 16 + l16;
      float v = acc[t][r] + bc[t] + X1[(size_t)rowC * DFEAT + c];
      vals[t] = v;
      s += v;
    }
    s += __shfl_xor(s, 1, 32); s += __shfl_xor(s, 2, 32);
    s += __shfl_xor(s, 4, 32); s += __shfl_xor(s, 8, 32);
    float mu = s * (1.0f / DFEAT);
    float q = 0.f;
#pragma unroll
    for (int t = 0; t < 8; ++t) { float d = vals[t] - mu; q += d * d; }
    q += __shfl_xor(q, 1, 32); q += __shfl_xor(q, 2, 32);
    q += __shfl_xor(q, 4, 32); q += __shfl_xor(q, 8, 32);
    float inv = rsqrtf(q * (1.0f / DFEAT) + 1e-5f);
    int rloc = r + half * 8;
#pragma unroll
    for (int t = 0; t < 8; ++t) {
      float xh = (vals[t] - mu) * inv * gl[t] + bl[t];
      lds[rloc * 132 + t * 16 + l16] = xh;       // normalized tile -> LDS
    }
  }
  __syncthreads();                               // single wave: S_NOP + waitcnts

  // ---- GEMM3: x2 @ Wsi^T, A-fragments from LDS
  v8f acc3[8];
#pragma unroll
  for (int t = 0; t < 8; ++t) acc3[t] = vzero;

#pragma unroll 4
  for (int kk = 0; kk < 32; ++kk) {
    const int k0 = kk * 4;
    v2f a = *reinterpret_cast<const v2f*>(&lds[l16 * 132 + k0 + kq]);
#pragma unroll
    for (int t = 0; t < 8; ++t) {
      const int colB = t * 16 + l16;
      v2f b = *reinterpret_cast<const v2f*>(Wsi + (size_t)colB * DFEAT + k0 + kq);
      acc3[t] = wmma4(a, b, acc3[t]);
    }
  }
  __syncthreads();

  // ---- ELU + residual + store (guards only around stores; WMMA all done)
#pragma unroll
  for (int r = 0; r < 8; ++r) {
    int rowR = i0 + r + half * 8;
    if (rowR < N) {
      int rloc = r + half * 8;
#pragma unroll
      for (int t = 0; t < 8; ++t) {
        int c = t * 16 + l16;
        float y = acc3[t][r] + bs[t];
        float e = y > 0.f ? y : (expf(y) - 1.f);
        out[(size_t)rowR * DFEAT + c] = e + lds[rloc * 132 + c];
      }
    }
  }
}

// ---------------------------------------------------------------- launch
extern "C" void kernel_launch(void* const* d_in, const int* in_sizes, int n_in,
                              void* d_out, int out_size, void* d_ws, size_t ws_size,
                              hipStream_t stream) {
  const float* h      = (const float*)d_in[0];
  const int*   src    = (const int*)d_in[1];
  const int*   dst    = (const int*)d_in[2];
  const float* Wself  = (const float*)d_in[3];
  const float* Wneigh = (const float*)d_in[4];
  const float* bconv  = (const float*)d_in[5];
  const float* g1     = (const float*)d_in[6];
  const float* b1     = (const float*)d_in[7];
  const float* g2     = (const float*)d_in[8];
  const float* b2     = (const float*)d_in[9];
  const float* Wsi    = (const float*)d_in[10];
  const float* bsi    = (const float*)d_in[11];
  float* out = (float*)d_out;

  const int N = in_sizes[0] / DFEAT;
  const int E = in_sizes[1];

  // workspace layout: h_norm[N*128] | agg[N*128] | deg[N]  (agg+deg contiguous)
  float* hnorm = (float*)d_ws;
  float* aggp  = hnorm + (size_t)N * DFEAT;
  float* degp  = aggp + (size_t)N * DFEAT;

  long nz = (long)N * DFEAT + N;     // zero agg + deg together
  zero_f32<<<(int)((nz + 255) / 256), 256, 0, stream>>>(aggp, nz);

  ln1_kernel<<<(N + 7) / 8, 256, 0, stream>>>(h, g1, b1, hnorm, N);

  deg_kernel<<<(E + 255) / 256, 256, 0, stream>>>(dst, degp, E);

  long sthreads = (long)E * 32;
  scatter_kernel<<<(int)((sthreads + 255) / 256), 256, 0, stream>>>(hnorm, src, dst, aggp, E);

  fused_conv_kernel<<<(N + 15) / 16, 32, 0, stream>>>(
      hnorm, aggp, degp, Wself, Wneigh, bconv, g2, b2, Wsi, bsi, out, N);
}